// MHARef_953482740143
// MI455X (gfx1250) — compile-verified
//
#include <hip/hip_runtime.h>
#include <cmath>

typedef __attribute__((ext_vector_type(2))) float v2f;
typedef __attribute__((ext_vector_type(4))) float v4f;
typedef __attribute__((ext_vector_type(8))) float v8f;

static __device__ __forceinline__ v8f wmma4(v2f a, v2f b, v8f c) {
  // V_WMMA_F32_16X16X4_F32 : D = A(16x4) x B(4x16) + C(16x16), fp32 throughout
  return __builtin_amdgcn_wmma_f32_16x16x4_f32(false, a, false, b, (short)0, c,
                                               false, false);
}

// ---------------------------------------------------------------------------
// C[m,n] = sum_k A[m,k] * W[n,k] (+ bias[n]).  A:[M,K] rm, W:[N,K] rm.
// Block = 128 threads = 4 waves, 64x64 tile. Double-buffered LDS staging of
// A and B k-chunks (KC=32): global prefetch overlaps the 32-WMMA compute.
// ---------------------------------------------------------------------------
__global__ __launch_bounds__(128) void gemm_xwT_wmma(
    const float* __restrict__ A, const float* __restrict__ W,
    const float* __restrict__ bias, float* __restrict__ C,
    int M, int N, int K) {
  constexpr int LP = 36;                   // padded LDS row stride (floats)
  __shared__ float lA[2][64 * LP];
  __shared__ float lB[2][64 * LP];

  const int tid  = threadIdx.x;
  const int lane = tid & 31;
  const int wave = tid >> 5;
  const int ml   = lane & 15;              // row (A) / col (B) within tile
  const int half = lane >> 4;              // 0: K={0,1}, 1: K={2,3}
  const int kb2  = half << 1;

  const int m0 = blockIdx.y * 64;
  const int n0 = blockIdx.x * 64;

  const float* Ablk = A + (size_t)m0 * K;
  const float* Wblk = W + (size_t)n0 * K;

  v4f ga[4], gb[4];
  auto stage = [&](const float* src, int kc, v4f* g) {   // coalesced b128
#pragma unroll
    for (int i = 0; i < 4; ++i) {
      const int idx = tid + i * 128;       // 64 rows x 8 float4
      const int row = idx >> 3;
      const int c4  = idx & 7;
      g[i] = *(const v4f*)(src + (size_t)row * K + kc + c4 * 4);
    }
  };
  auto commit = [&](float* lds, const v4f* g) {
#pragma unroll
    for (int i = 0; i < 4; ++i) {
      const int idx = tid + i * 128;
      const int row = idx >> 3;
      const int c4  = idx & 7;
      *(v4f*)(lds + row * LP + c4 * 4) = g[i];
    }
  };

  stage(Ablk, 0, ga);
  stage(Wblk, 0, gb);
  commit(lA[0], ga);
  commit(lB[0], gb);

  v8f acc0 = {}, acc1 = {}, acc2 = {}, acc3 = {};
  const int nc = K >> 5;                   // K / 32 chunks
  for (int c = 0; c < nc; ++c) {
    __syncthreads();                       // chunk c visible; buf c+1 free
    if (c + 1 < nc) {                      // prefetch next chunk (no wait yet)
      stage(Ablk, (c + 1) << 5, ga);
      stage(Wblk, (c + 1) << 5, gb);
    }
    const float* la = &lA[c & 1][(wave * 16 + ml) * LP + kb2];
    const float* lb = &lB[c & 1][ml * LP + kb2];
#pragma unroll
    for (int ks = 0; ks < 8; ++ks) {       // 8 k-steps x 4 WMMA
      v2f a  = *(const v2f*)(la + ks * 4);
      v2f b0 = *(const v2f*)(lb + ks * 4);
      v2f b1 = *(const v2f*)(lb + 16 * LP + ks * 4);
      v2f b2 = *(const v2f*)(lb + 32 * LP + ks * 4);
      v2f b3 = *(const v2f*)(lb + 48 * LP + ks * 4);
      acc0 = wmma4(a, b0, acc0);
      acc1 = wmma4(a, b1, acc1);
      acc2 = wmma4(a, b2, acc2);
      acc3 = wmma4(a, b3, acc3);
    }
    if (c + 1 < nc) {                      // loadcnt wait lands here, post-math
      commit(lA[(c + 1) & 1], ga);
      commit(lB[(c + 1) & 1], gb);
    }
  }

  v8f accs[4] = {acc0, acc1, acc2, acc3};
#pragma unroll
  for (int t = 0; t < 4; ++t) {
    const int col  = n0 + t * 16 + ml;
    const float bv = bias ? bias[col] : 0.0f;
#pragma unroll
    for (int r = 0; r < 8; ++r) {
      const int row = m0 + wave * 16 + r + 8 * half;
      C[(size_t)row * N + col] = accs[t][r] + bv;
    }
  }
}

// ---------------------------------------------------------------------------
// Per (b,h): scores = (Q K^T)*8 with causal mask, softmax -> attn weights
// (written to output), ctx = P V.  qkv layout: [b, s, {q,k,v}, h, dh].
// One wave per 16-row query tile; K-tiles double-buffered in registers.
// ---------------------------------------------------------------------------
__global__ __launch_bounds__(128) void attn_softmax_wmma(
    const float* __restrict__ qkv, float* __restrict__ attn,
    float* __restrict__ ctx) {
  __shared__ float s_max[4][16];
  __shared__ float s_sum[4][16];

  const int lane = threadIdx.x & 31;
  const int wave = threadIdx.x >> 5;
  const int ml   = lane & 15;
  const int half = lane >> 4;
  const int kb2  = half << 1;

  const int bh = blockIdx.x;               // B*H = 64
  const int b  = bh >> 4;
  const int h  = bh & 15;
  const int qt = blockIdx.y * 4 + wave;    // query tile 0..63
  const int q0 = qt * 16;

  const int S = 1024;
  const size_t tokStride = 3072;           // floats per token in qkv
  const float* base = qkv + (size_t)b * S * tokStride + (size_t)h * 64;

  // Q block [16 x 64] in registers, WMMA A-layout (16 k-steps of 4)
  v2f qa[16];
  {
    const float* qr = base + (size_t)(q0 + ml) * tokStride + kb2;
#pragma unroll
    for (int ks = 0; ks < 16; ++ks) qa[ks] = *(const v2f*)(qr + ks * 4);
  }

  float* arow = attn + ((size_t)bh * S + q0) * S;

  float rmax[8], rsum[8];
#pragma unroll
  for (int r = 0; r < 8; ++r) { rmax[r] = -__builtin_inff(); rsum[r] = 0.0f; }

  auto loadK = [&](int j, v2f* kb) {
    const float* kr = base + (size_t)(j * 16 + ml) * tokStride + 1024 + kb2;
#pragma unroll
    for (int ks = 0; ks < 16; ++ks) kb[ks] = *(const v2f*)(kr + ks * 4);
  };

  auto scoreTile = [&](int j, const v2f* kb) {
    v8f acc = {};
#pragma unroll
    for (int ks = 0; ks < 16; ++ks) acc = wmma4(qa[ks], kb[ks], acc);
    const bool diag = (j == qt);
#pragma unroll
    for (int r = 0; r < 8; ++r) {
      const int lrow = r + 8 * half;       // local row in tile
      const int col  = j * 16 + ml;        // global key index
      float sc = acc[r] * 8.0f;            // faithful bug: * sqrt(DH)
      if (diag && col > q0 + lrow) sc = -__builtin_inff();
      arow[(size_t)lrow * S + col] = sc;   // raw (pre-softmax) store

      float tm = sc;                       // row-max over 16 lanes in half
      tm = fmaxf(tm, __shfl_xor(tm, 1));
      tm = fmaxf(tm, __shfl_xor(tm, 2));
      tm = fmaxf(tm, __shfl_xor(tm, 4));
      tm = fmaxf(tm, __shfl_xor(tm, 8));
      const float mnew = fmaxf(rmax[r], tm);
      float e = __expf(sc - mnew);         // -inf -> 0
      e += __shfl_xor(e, 1);
      e += __shfl_xor(e, 2);
      e += __shfl_xor(e, 4);
      e += __shfl_xor(e, 8);
      const float corr =
          (rmax[r] == -__builtin_inff()) ? 0.0f : __expf(rmax[r] - mnew);
      rsum[r] = rsum[r] * corr + e;
      rmax[r] = mnew;
    }
  };

  // ---- Pass 1: double-buffered K tiles, load j+1 while WMMA-ing j ----
  v2f kb0[16], kb1[16];
  loadK(0, kb0);
  for (int j = 0; j <= qt; j += 2) {
    if (j + 1 <= qt) loadK(j + 1, kb1);
    scoreTile(j, kb0);
    if (j + 1 <= qt) {
      if (j + 2 <= qt) loadK(j + 2, kb0);
      scoreTile(j + 1, kb1);
    }
  }

  // publish per-row stats so A-layout lanes (row = ml) can grab them
  if (ml == 0) {
#pragma unroll
    for (int r = 0; r < 8; ++r) {
      s_max[wave][r + 8 * half] = rmax[r];
      s_sum[wave][r + 8 * half] = rsum[r];
    }
  }
  __syncthreads();

  const float myMax = s_max[wave][ml];
  const float myInv = 1.0f / s_sum[wave][ml];

  // raw scores are re-read by *different* lanes of this wave: drain stores
  asm volatile("s_wait_storecnt 0x0" ::: "memory");

  // ---- Pass 2: normalize weights in-place, ctx = P x V via WMMA ----
  v8f c0 = {}, c1 = {}, c2 = {}, c3 = {};
  const float* vbase = base + 2048;                   // V matrix
  float* prow = arow + (size_t)ml * S + kb2;          // A-layout row

  for (int j = 0; j < 64; ++j) {
    if (j <= qt) {
      v2f p[4];
      v2f bb[16];                                     // [ks][t]
#pragma unroll
      for (int ks = 0; ks < 4; ++ks) {                // gather everything
        const int cb = j * 16 + ks * 4;
        v2f t2 = *(v2f*)(prow + cb);
        t2.x = __expf(t2.x - myMax) * myInv;
        t2.y = __expf(t2.y - myMax) * myInv;
        *(v2f*)(prow + cb) = t2;                      // final attn weight
        p[ks] = t2;
        const float* v0 = vbase + (size_t)(cb + kb2) * tokStride;
        const float* v1 = v0 + tokStride;
#pragma unroll
        for (int t = 0; t < 4; ++t) {
          v2f bv;
          bv.x = v0[t * 16 + ml];
          bv.y = v1[t * 16 + ml];
          bb[ks * 4 + t] = bv;
        }
      }
#pragma unroll
      for (int ks = 0; ks < 4; ++ks) {                // then 16 WMMAs
        c0 = wmma4(p[ks], bb[ks * 4 + 0], c0);
        c1 = wmma4(p[ks], bb[ks * 4 + 1], c1);
        c2 = wmma4(p[ks], bb[ks * 4 + 2], c2);
        c3 = wmma4(p[ks], bb[ks * 4 + 3], c3);
      }
    } else {  // fully-masked tile: attn weight is exactly 0
      v2f z = {0.0f, 0.0f};
#pragma unroll
      for (int ks = 0; ks < 4; ++ks) *(v2f*)(prow + j * 16 + ks * 4) = z;
    }
  }

  // ctx[b, s, h*64+dh]  (C/D layout: VGPR r -> row r+8*half, col = ml)
  v8f cs[4] = {c0, c1, c2, c3};
  float* cbase = ctx + (size_t)b * S * 1024 + (size_t)h * 64;
#pragma unroll
  for (int t = 0; t < 4; ++t) {
    const int dh = t * 16 + ml;
#pragma unroll
    for (int r = 0; r < 8; ++r) {
      const int row = q0 + r + 8 * half;
      cbase[(size_t)row * 1024 + dh] = cs[t][r];
    }
  }
}

// ---------------------------------------------------------------------------
extern "C" void kernel_launch(void* const* d_in, const int* in_sizes, int n_in,
                              void* d_out, int out_size, void* d_ws,
                              size_t ws_size, hipStream_t stream) {
  (void)in_sizes; (void)n_in; (void)out_size; (void)ws_size;
  const float* x      = (const float*)d_in[0];
  // d_in[1]=y, d_in[2]=z, d_in[3]=key_padding_mask : unused by reference
  const float* W_qkv  = (const float*)d_in[4];
  const float* W_proj = (const float*)d_in[5];
  const float* b_proj = (const float*)d_in[6];

  const int Bb = 4, S = 1024, D = 1024;
  float* out  = (float*)d_out;
  float* attn = out + (size_t)Bb * S * D;            // tuple: (out, attn)
  float* qkvb = (float*)d_ws;                        // 48 MB
  float* ctxb = qkvb + (size_t)Bb * S * 3 * D;       // +16 MB

  // 1) qkv = x @ W_qkv^T      [4096 x 3072], K=1024
  gemm_xwT_wmma<<<dim3(3072 / 64, 4096 / 64), 128, 0, stream>>>(
      x, W_qkv, nullptr, qkvb, Bb * S, 3 * D, D);

  // 2) attention: scores -> masked/scaled softmax -> attn weights + ctx
  attn_softmax_wmma<<<dim3(64, 16), 128, 0, stream>>>(qkvb, attn, ctxb);

  // 3) out = ctx @ W_proj^T + b_proj   [4096 x 1024], K=1024
  gemm_xwT_wmma<<<dim3(1024 / 64, 4096 / 64), 128, 0, stream>>>(
      ctxb, W_proj, b_proj, out, Bb * S, D, D);
}